// VectorQuantizer_76991583748601
// MI455X (gfx1250) — compile-verified
//
#include <hip/hip_runtime.h>

// VQ quantize: x (16,64,64,64) f32, codebook e (64,1024) f32 -> out (16,64,64,64) f32
// score_{n,k} = x_n . e_k - 0.5*||e_k||^2 ;  out row = e[:, argmax_k score]

typedef float v2f __attribute__((ext_vector_type(2)));
typedef float v8f __attribute__((ext_vector_type(8)));

#define DIMS  64
#define NK    1024
#define HW    4096
#define CHW   (64 * 4096)
#define NPTS  65536          // 16 * 64 * 64

// ---------------------------------------------------------------------------
// Kernel 1: per-codebook-column bias = -0.5 * ||e_k||^2
// ---------------------------------------------------------------------------
__global__ void vq_norms_kernel(const float* __restrict__ e, float* __restrict__ nrm) {
    int k = blockIdx.x * blockDim.x + threadIdx.x;
    if (k < NK) {
        float s = 0.f;
#pragma unroll
        for (int d = 0; d < DIMS; ++d) {
            float v = e[d * NK + k];
            s += v * v;
        }
        nrm[k] = -0.5f * s;
    }
}

// ---------------------------------------------------------------------------
// Kernel 2: swizzle codebook into WMMA B-fragment lane order.
//   For k-tile kt, chunk c, lane L (32 lanes):
//     d0 = 4c + 2*(L>>4), k = kt*16 + (L&15)
//     ebuf[kt*1024 + c*64 + 2L + {0,1}] = e[d0*NK + k], e[(d0+1)*NK + k]
//   Hot loop then does one coalesced b64 load per chunk.
// ---------------------------------------------------------------------------
__global__ void vq_pack_kernel(const float* __restrict__ e, float* __restrict__ ebuf) {
    int tid = blockIdx.x * blockDim.x + threadIdx.x;   // 0 .. 32767
    int L  = tid & 31;
    int c  = (tid >> 5) & 15;
    int kt = tid >> 9;
    int d0 = 4 * c + 2 * (L >> 4);
    int k  = kt * 16 + (L & 15);
    v2f v;
    v.x = e[d0 * NK + k];
    v.y = e[(d0 + 1) * NK + k];
    *(v2f*)(ebuf + (size_t)tid * 2) = v;
}

// ---------------------------------------------------------------------------
// Kernel 3: WMMA score matmul + argmax + gather/store
//   block = 256 threads = 8 waves; wave handles 32 points (two 16x16 M-tiles)
//   grid  = 65536 / 256 = 256 blocks
// ---------------------------------------------------------------------------
__global__ __launch_bounds__(256) void vq_main_kernel(const float* __restrict__ x,
                                                      const float* __restrict__ e,
                                                      const float* __restrict__ nrm,
                                                      const float* __restrict__ ebuf,
                                                      float* __restrict__ out) {
    __shared__ int kbest_lds[8 * 32];

    const int lane = threadIdx.x & 31;
    const int wave = threadIdx.x >> 5;
    const int l15  = lane & 15;         // column / row-in-tile position
    const int half = lane >> 4;         // 0: lanes 0-15, 1: lanes 16-31
    const int base = (blockIdx.x * 8 + wave) * 32;   // first point of this wave
    const int b    = base >> 12;        // batch index (HW = 4096)
    const int hw   = base & 4095;       // h*W + w of first point (same b for all 32)

    // ---- Load A tiles into registers -------------------------------------
    // A 16x4 f32 layout: lane<16 holds K=4c,4c+1 ; lane>=16 holds K=4c+2,4c+3 ; M = lane&15
    v2f a[2][16];
#pragma unroll
    for (int t = 0; t < 2; ++t) {
        const float* xp = x + (size_t)b * CHW + (size_t)(hw + t * 16 + l15);
#pragma unroll
        for (int c = 0; c < 16; ++c) {
            const int d0 = 4 * c + 2 * half;
            v2f av;
            av.x = xp[(size_t)d0 * HW];
            av.y = xp[(size_t)(d0 + 1) * HW];
            a[t][c] = av;
        }
    }

    float best[2][8];
    int   bidx[2][8];
#pragma unroll
    for (int t = 0; t < 2; ++t)
#pragma unroll
        for (int v = 0; v < 8; ++v) { best[t][v] = -3.4e38f; bidx[t][v] = 0; }

    // ---- Sweep the 64 k-tiles --------------------------------------------
    const float* ebl = ebuf + (size_t)lane * 2;   // lane's slot in swizzled B
    for (int kt = 0; kt < 64; ++kt) {
        const int kcol = kt * 16 + l15;             // this lane's column N
        const float cinit = nrm[kcol];              // -0.5*||e_k||^2 bias

        v8f acc0, acc1;
#pragma unroll
        for (int v = 0; v < 8; ++v) { acc0[v] = cinit; acc1[v] = cinit; }

        const float* ep = ebl + (size_t)kt * 1024;
#pragma unroll
        for (int c = 0; c < 16; ++c) {
            v2f bv = *(const v2f*)(ep + c * 64);    // coalesced b64, imm offset
            acc0 = __builtin_amdgcn_wmma_f32_16x16x4_f32(false, a[0][c], false, bv,
                                                         (short)0, acc0, false, false);
            acc1 = __builtin_amdgcn_wmma_f32_16x16x4_f32(false, a[1][c], false, bv,
                                                         (short)0, acc1, false, false);
        }

#pragma unroll
        for (int v = 0; v < 8; ++v) {
            if (acc0[v] > best[0][v]) { best[0][v] = acc0[v]; bidx[0][v] = kcol; }
            if (acc1[v] > best[1][v]) { best[1][v] = acc1[v]; bidx[1][v] = kcol; }
        }
    }

    // ---- Cross-lane argmax (within each 16-lane half; row set is per-half) -
#pragma unroll
    for (int m = 1; m <= 8; m <<= 1) {
#pragma unroll
        for (int t = 0; t < 2; ++t)
#pragma unroll
            for (int v = 0; v < 8; ++v) {
                float ov = __shfl_xor(best[t][v], m, 32);
                int   oi = __shfl_xor(bidx[t][v], m, 32);
                // max score; tie -> lower index (matches argmin-first semantics)
                if (ov > best[t][v] || (ov == best[t][v] && oi < bidx[t][v])) {
                    best[t][v] = ov;
                    bidx[t][v] = oi;
                }
            }
    }

    // ---- Publish winners: point offset q = t*16 + (v + 8*half) -----------
#pragma unroll
    for (int v = 0; v < 8; ++v) {
        if (l15 == v) {
            kbest_lds[wave * 32 + 0 * 16 + v + 8 * half] = bidx[0][v];
            kbest_lds[wave * 32 + 1 * 16 + v + 8 * half] = bidx[1][v];
        }
    }
    __syncthreads();

    // ---- Gather codebook rows & coalesced store (lane == point) ----------
    const int k0 = kbest_lds[wave * 32 + lane];
    const size_t obase = (size_t)b * CHW + (size_t)(hw + lane);
#pragma unroll 8
    for (int c = 0; c < DIMS; ++c) {
        out[obase + (size_t)c * HW] = e[c * NK + k0];
    }
}

// ---------------------------------------------------------------------------
extern "C" void kernel_launch(void* const* d_in, const int* in_sizes, int n_in,
                              void* d_out, int out_size, void* d_ws, size_t ws_size,
                              hipStream_t stream) {
    const float* x = (const float*)d_in[0];     // (16, 64, 64, 64)
    const float* e = (const float*)d_in[1];     // (64, 1024)
    float* out  = (float*)d_out;                // (16, 64, 64, 64)
    float* nrm  = (float*)d_ws;                 // 1024 floats
    float* ebuf = (float*)d_ws + 1024;          // 65536 floats, swizzled B fragments

    vq_norms_kernel<<<NK / 256, 256, 0, stream>>>(e, nrm);
    vq_pack_kernel<<<(64 * 16 * 32) / 256, 256, 0, stream>>>(e, ebuf);
    vq_main_kernel<<<NPTS / 256, 256, 0, stream>>>(x, e, nrm, ebuf, out);
}